// ParallelLinear_59133109732019
// MI455X (gfx1250) — compile-verified
//
#include <hip/hip_runtime.h>

// CDNA5 / gfx1250: wave32, WMMA f32 16x16x4.
typedef __attribute__((ext_vector_type(2))) float v2f;
typedef __attribute__((ext_vector_type(4))) float v4f;
typedef __attribute__((ext_vector_type(8))) float v8f;

#define T_DIM 1024
#define ROWS_PER_BLOCK 32
#define LDS_STRIDE 68   // 64 + 4 pad: bank = (m*68 + c) % 64 = 4m + c -> conflict-free fragment reads

__global__ __launch_bounds__(256)
void ParallelLinear_59133109732019_kernel(const float* __restrict__ x,
                                          const float* __restrict__ W,
                                          const float* __restrict__ bias,
                                          float* __restrict__ out) {
  __shared__ float sW[64 * LDS_STRIDE];             // W_t, row o, 64 cols (i), padded
  __shared__ float sX[ROWS_PER_BLOCK * LDS_STRIDE]; // X tile, 32 rows, 64 cols (i), padded
  __shared__ float sB[64];                          // bias_t

  const int t = blockIdx.x;                 // time slice: its own 64x64 weight
  const int rowbase = blockIdx.y * ROWS_PER_BLOCK;
  const int tid = threadIdx.x;

  // ---- Stage W_t (64x64 fp32 = 16KB) into LDS, coalesced b128 (keep in L2: reused 16x) ----
  const v4f* Wg = (const v4f*)(W + (size_t)t * 64 * 64);
  #pragma unroll
  for (int it = 0; it < 4; ++it) {
    int j  = tid + it * 256;          // 0..1023 float4s
    int r  = j >> 4;                  // row 0..63
    int c4 = j & 15;                  // float4 col 0..15
    *(v4f*)&sW[r * LDS_STRIDE + c4 * 4] = Wg[j];
  }
  // ---- Stage 32x64 X tile (8KB), non-temporal (each x element read once) ----
  #pragma unroll
  for (int it = 0; it < 2; ++it) {
    int j  = tid + it * 256;          // 0..511 float4s
    int r  = j >> 4;                  // row 0..31
    int c4 = j & 15;
    const v4f* Xg = (const v4f*)(x + ((size_t)(rowbase + r) * T_DIM + t) * 64);
    *(v4f*)&sX[r * LDS_STRIDE + c4 * 4] = __builtin_nontemporal_load(Xg + c4);
  }
  if (tid < 64) sB[tid] = bias[(size_t)t * 64 + tid];
  __syncthreads();

  // ---- 8 waves: 2 M-tiles x 4 N-tiles of 16x16 outputs ----
  const int lane  = tid & 31;
  const int wave  = tid >> 5;
  const int mtile = wave >> 2;        // 0..1
  const int ntile = wave & 3;         // 0..3
  const int mn    = lane & 15;        // M index for A-frag, N index for B/C/D-frag
  const int half  = lane >> 4;        // K-half selector for A/B frags, M-half for C/D

  // Seed accumulator with bias (depends only on N == lane&15)
  v8f acc;
  const float bv = sB[ntile * 16 + mn];
  #pragma unroll
  for (int j = 0; j < 8; ++j) acc[j] = bv;

  // A frag (16x4 f32): lane holds X[m = lane&15][k0 + 2*half + {0,1}]
  // B frag (4x16 f32): B[k][n] = W[n][k] -> lane holds W[ntile*16 + (lane&15)][k0 + 2*half + {0,1}]
  const float* aRow = &sX[(mtile * 16 + mn) * LDS_STRIDE + 2 * half];
  const float* bRow = &sW[(ntile * 16 + mn) * LDS_STRIDE + 2 * half];

  #pragma unroll
  for (int k0 = 0; k0 < 64; k0 += 4) {
    v2f a  = *(const v2f*)(aRow + k0);   // ds_load_b64, bank-conflict-free
    v2f bb = *(const v2f*)(bRow + k0);
    // 8-arg form: (neg_a, A, neg_b, B, c_mod, C, reuse_a, reuse_b)
    acc = __builtin_amdgcn_wmma_f32_16x16x4_f32(false, a, false, bb,
                                                (short)0, acc, false, false);
  }

  // ---- Store D: VGPR j holds (M = j + 8*half, N = lane&15); write-once -> non-temporal ----
  const int ncol = ntile * 16 + mn;
  #pragma unroll
  for (int j = 0; j < 8; ++j) {
    int row = rowbase + mtile * 16 + 8 * half + j;  // global B index
    __builtin_nontemporal_store(acc[j], out + ((size_t)row * T_DIM + t) * 64 + ncol);
  }
}

extern "C" void kernel_launch(void* const* d_in, const int* in_sizes, int n_in,
                              void* d_out, int out_size, void* d_ws, size_t ws_size,
                              hipStream_t stream) {
  (void)in_sizes; (void)n_in; (void)out_size; (void)d_ws; (void)ws_size;
  const float* x  = (const float*)d_in[0];   // [512,1024,8,8] == [B,T,64]
  const float* W  = (const float*)d_in[1];   // [1024,64,64]
  const float* b  = (const float*)d_in[2];   // [1024,64]
  float* out      = (float*)d_out;           // [512,1024,64]

  dim3 grid(T_DIM, 512 / ROWS_PER_BLOCK);    // (1024, 16)
  ParallelLinear_59133109732019_kernel<<<grid, 256, 0, stream>>>(x, W, b, out);
}